// ROIPoolingLayer_52750788329711
// MI455X (gfx1250) — compile-verified
//
#include <hip/hip_runtime.h>
#include <stdint.h>

// ---------------------------------------------------------------------------
// ROI-Align (crop_and_resize, bilinear, extrapolation=0) for MI455X (gfx1250).
// feature_map: [8,64,64,256] f32 (channels innermost), rois: [8,512,4] f32
// (xc,yc,w,h), out: [8,512,7,7,256] fp16.
//
// Bandwidth-bound (~136MB HBM @ 23.3TB/s ~ 6us). Uses CDNA5 async
// global->LDS DMA (GLOBAL_LOAD_ASYNC_TO_LDS_B128 + ASYNCcnt) to stage the
// 4 bilinear-corner channel vectors per output cell, double-buffered per wave.
// ---------------------------------------------------------------------------

typedef float    f32x4 __attribute__((ext_vector_type(4)));
typedef uint32_t u32x4 __attribute__((ext_vector_type(4)));

#define NWAVES 4
#define BLOCK  (NWAVES * 32)

// Async DMA: copies 16 bytes per lane from the global address g to the LDS address lds_byte.
// Tracked by ASYNCcnt, not LOADcnt.
__device__ __forceinline__ void async_ld_b128(uint32_t lds_byte, const float* g) {
    uint64_t ga = (uint64_t)(uintptr_t)g;
    asm volatile("global_load_async_to_lds_b128 %0, %1, off"
                 :: "v"(lds_byte), "v"(ga)
                 : "memory");
}
__device__ __forceinline__ void wait_async_8() {
    asm volatile("s_wait_asynccnt 8" ::: "memory");
}
__device__ __forceinline__ void wait_async_0() {
    asm volatile("s_wait_asynccnt 0" ::: "memory");
}

__global__ __launch_bounds__(BLOCK)
void roi_align_gfx1250(const float* __restrict__ fm,
                       const float* __restrict__ rois,
                       _Float16* __restrict__ out)
{
    // [wave][buf][corner(tl,tr,bl,br)][channel]
    __shared__ alignas(16) float lds[NWAVES][2][4][256];

    const int roiIdx = blockIdx.x;        // b*512 + r
    const int b      = roiIdx >> 9;
    const int wave   = threadIdx.x >> 5;
    const int lane   = threadIdx.x & 31;

    const f32x4 roi = ((const f32x4*)rois)[roiIdx];   // xc, yc, w, h
    const float xc = roi.x, yc = roi.y, rw = roi.z, rh = roi.w;

    // ys = (yc - h/2)*63 + iy * h*63/6 ; same for x. (H=W=64, PH=PW=7)
    const float y1 = (yc - 0.5f * rh) * 63.0f;
    const float x1 = (xc - 0.5f * rw) * 63.0f;
    const float dy = rh * (63.0f / 6.0f);
    const float dx = rw * (63.0f / 6.0f);

    const float* fbase = fm + (size_t)b * (64 * 64 * 256);

    auto issue = [&](int cell, int buf) {
        const int   iy = cell / 7, ix = cell % 7;
        const float ys = y1 + (float)iy * dy;
        const float xs = x1 + (float)ix * dx;
        // Clamped corners: safe addresses even for out-of-bounds (masked) cells.
        // floor/floor+1 matches floor/ceil since lerp weight is 0 when they differ.
        int yt = (int)floorf(ys); yt = yt < 0 ? 0 : (yt > 63 ? 63 : yt);
        int xl = (int)floorf(xs); xl = xl < 0 ? 0 : (xl > 63 ? 63 : xl);
        const int yb = yt + 1 > 63 ? 63 : yt + 1;
        const int xr = xl + 1 > 63 ? 63 : xl + 1;

        const int pix[4] = { yt * 64 + xl, yt * 64 + xr,
                             yb * 64 + xl, yb * 64 + xr };
        const uint32_t lbase =
            (uint32_t)(uintptr_t)(&lds[wave][buf][0][0]);  // low 32 bits of the pointer are the LDS offset
#pragma unroll
        for (int k = 0; k < 4; ++k) {
            const float*   g = fbase + (size_t)pix[k] * 256 + lane * 4;
            const uint32_t l = lbase + (uint32_t)((k * 256 + lane * 4) * 4);
            async_ld_b128(l,       g);        // channels [lane*4 ..)
            async_ld_b128(l + 512, g + 128);  // channels [128+lane*4 ..)
        }
    };

    auto compute_store = [&](int cell, int buf) {
        const int   iy = cell / 7, ix = cell % 7;
        const float ys = y1 + (float)iy * dy;
        const float xs = x1 + (float)ix * dx;
        const bool  ok = (ys >= 0.0f) && (ys <= 63.0f) &&
                         (xs >= 0.0f) && (xs <= 63.0f);
        const float vy  = ys - floorf(ys);
        const float vx  = xs - floorf(xs);
        const float msk = ok ? 1.0f : 0.0f;

        const f32x4* L = (const f32x4*)&lds[wave][buf][0][0];  // 64 vec4 / corner

        auto lerp = [&](float a, float bb, float c, float d) -> float {
            const float top = a + (bb - a) * vx;
            const float bot = c + (d  - c) * vx;
            return (top + (bot - top) * vy) * msk;
        };

        _Float16 hv[8];
#pragma unroll
        for (int j = 0; j < 2; ++j) {
            const int   i  = lane * 2 + j;
            const f32x4 tl = L[i];
            const f32x4 tr = L[64  + i];
            const f32x4 bl = L[128 + i];
            const f32x4 br = L[192 + i];
            hv[j * 4 + 0] = (_Float16)lerp(tl.x, tr.x, bl.x, br.x);
            hv[j * 4 + 1] = (_Float16)lerp(tl.y, tr.y, bl.y, br.y);
            hv[j * 4 + 2] = (_Float16)lerp(tl.z, tr.z, bl.z, br.z);
            hv[j * 4 + 3] = (_Float16)lerp(tl.w, tr.w, bl.w, br.w);
        }
        u32x4 bits;
        __builtin_memcpy(&bits, hv, 16);
        _Float16* op = out + ((size_t)roiIdx * 49 + cell) * 256 + lane * 8;
        __builtin_nontemporal_store(bits, (u32x4*)op);  // write-once; keep L2 for fm
    };

    // Software pipeline over this wave's cells (49 cells round-robin over 4 waves).
    int cell = wave;
    int buf  = 0;
    issue(cell, buf);
    for (;;) {
        const int next = cell + NWAVES;
        if (next < 49) {
            issue(next, buf ^ 1);   // 8 more async loads in flight
            wait_async_8();         // cell's 8 loads have landed (in-order)
        } else {
            wait_async_0();
        }
        compute_store(cell, buf);
        if (next >= 49) break;
        cell = next;
        buf ^= 1;
    }
}

extern "C" void kernel_launch(void* const* d_in, const int* in_sizes, int n_in,
                              void* d_out, int out_size, void* d_ws, size_t ws_size,
                              hipStream_t stream) {
    const float* fm   = (const float*)d_in[0];   // [8,64,64,256] f32
    const float* rois = (const float*)d_in[1];   // [8,512,4]    f32
    _Float16*    out  = (_Float16*)d_out;        // [8,512,7,7,256] fp16

    dim3 grid(8 * 512);   // one block per ROI
    dim3 block(BLOCK);
    hipLaunchKernelGGL(roi_align_gfx1250, grid, block, 0, stream, fm, rois, out);
}